// Graph_ODE_RNN_2705829397223
// MI455X (gfx1250) — compile-verified
//
#include <hip/hip_runtime.h>
#include <math.h>

// CDNA5 (gfx1250) graph-encoder pipeline.
// GEMM work (pool, edge messages, head) on V_WMMA_F32_16X16X4_F32 (fp32 pipe,
// matches fp32 reference). wave32; A-operands stored K-contiguous for b64
// loads; branch-free masked B loads; pool does 4 output tiles per wave.

typedef float v2f __attribute__((ext_vector_type(2)));
typedef float v8f __attribute__((ext_vector_type(8)));

#if defined(__HIP_DEVICE_COMPILE__)
#if !__has_builtin(__builtin_amdgcn_wmma_f32_16x16x4_f32)
#error "device: __builtin_amdgcn_wmma_f32_16x16x4_f32 not available"
#endif
#endif

static __device__ __forceinline__ v8f wmma_k4(v2f a, v2f b, v8f c) {
#if defined(__HIP_DEVICE_COMPILE__)
  // (neg_a, A, neg_b, B, c_mod, C, reuse_a, reuse_b)
  return __builtin_amdgcn_wmma_f32_16x16x4_f32(false, a, false, b, (short)0, c,
                                               false, false);
#else
  (void)a; (void)b;
  return c;  // host-pass stub
#endif
}

#define V8F_ZERO {0.f, 0.f, 0.f, 0.f, 0.f, 0.f, 0.f, 0.f}

// ---------------------------------------------------------------- utilities

__global__ void fill_zero_kernel(float* __restrict__ p, long long n) {
  long long i = (long long)blockIdx.x * blockDim.x + threadIdx.x;
  long long stride = (long long)gridDim.x * blockDim.x;
  for (; i < n; i += stride) p[i] = 0.0f;
}

// Edge basis weights, stored TRANSPOSED: WeT[e, o, i] so the WMMA A-operand
// (K = Cin) is contiguous.  WeT[e,o,i] = sum_k softmax(ea[e] @ A)[k]*Wk[k,i,o]
__global__ void edgew_kernel(const float* __restrict__ ea,
                             const float* __restrict__ A,
                             const float* __restrict__ Wk,
                             float* __restrict__ WeT, int E, int Cin,
                             int Cout) {
  int CiCo = Cin * Cout;
  long long idx = (long long)blockIdx.x * blockDim.x + threadIdx.x;
  if (idx >= (long long)E * CiCo) return;
  int e = (int)(idx / CiCo);
  int j = (int)(idx % CiCo);
  int o = j / Cin, i = j % Cin;
  int w = i * Cout + o;  // flat index into one Wk[k] slab
  const float* a = ea + (size_t)e * 3;
  float z0 = a[0] * A[0] + a[1] * A[3] + a[2] * A[6];
  float z1 = a[0] * A[1] + a[1] * A[4] + a[2] * A[7];
  float z2 = a[0] * A[2] + a[1] * A[5] + a[2] * A[8];
  float mx = fmaxf(z0, fmaxf(z1, z2));
  float p0 = __expf(z0 - mx), p1 = __expf(z1 - mx), p2 = __expf(z2 - mx);
  float inv = 1.0f / (p0 + p1 + p2);
  WeT[idx] = (p0 * Wk[w] + p1 * Wk[CiCo + w] + p2 * Wk[2 * CiCo + w]) * inv;
}

// WT[o, k] = W[k, o]  (make head-weight A-operand K-contiguous)
__global__ void transpose_w_kernel(const float* __restrict__ W,
                                   float* __restrict__ WT, int Cin, int Cout) {
  int idx = blockIdx.x * blockDim.x + threadIdx.x;
  if (idx >= Cin * Cout) return;
  int o = idx / Cin, k = idx % Cin;
  WT[idx] = W[k * Cout + o];
}

__global__ void count_kernel(const int* __restrict__ ei,
                             float* __restrict__ cnt, int E) {
  int e = blockIdx.x * blockDim.x + threadIdx.x;
  if (e >= E) return;
  atomicAdd(&cnt[ei[E + e]], 1.0f);
}

// Level-0 messages: Cin == 1 -> rank-1 outer product, VALU + atomics.
__global__ void msg1_kernel(const float* __restrict__ x,
                            const float* __restrict__ We,  // [E,16] (T==same)
                            const int* __restrict__ ei, float* __restrict__ agg,
                            int N, int E, int Cout, int T, long long total) {
  long long idx = (long long)blockIdx.x * blockDim.x + threadIdx.x;
  if (idx >= total) return;
  int t = (int)(idx % T);
  int e = (int)((idx / T) % E);
  int b = (int)(idx / ((long long)T * E));
  int src = ei[e], dst = ei[E + e];
  float xv = x[((size_t)(b * N + src)) * T + t];
  float* base = agg + ((size_t)(b * N + dst) * Cout) * T + t;
  const float* w = We + (size_t)e * Cout;
  for (int o = 0; o < Cout; ++o)
    atomicAdd(&base[(size_t)o * T], w[o] * xv);
}

// Edge-message GEMM: C[16x16] = WeT[e] (16 x Cin, K-contig) @ x[b,src]
// (Cin x 16), atomically scattered into agg[b,dst].
__global__ void msg_wmma_kernel(const float* __restrict__ x,    // [B,N,Cin,T]
                                const float* __restrict__ WeT,  // [E,Cout,Cin]
                                const int* __restrict__ ei,     // [2,E]
                                float* __restrict__ agg,        // [B,N,Cout,T]
                                int N, int E, int Cin, int Cout, int B, int T,
                                int numTiles) {
  int wave = blockIdx.x * (blockDim.x >> 5) + (threadIdx.x >> 5);
  if (wave >= numTiles) return;
  int lane = threadIdx.x & 31;
  int tTiles = (T + 15) >> 4;
  int oTiles = Cout >> 4;
  int tt = wave % tTiles;
  int tmp = wave / tTiles;
  int ot = tmp % oTiles;
  tmp /= oTiles;
  int b = tmp % B;
  int e = tmp / B;
  int src = ei[e], dst = ei[E + e];

  int m = lane & 15;   // A row / C column base
  int kh = lane >> 4;  // K half-select
  int o = (ot << 4) + m;
  int t0 = (tt << 4) + m;
  float msk = t0 < T ? 1.0f : 0.0f;  // branch-free tail masking
  int tcl = t0 < T ? t0 : T - 1;     // clamped (always-valid) address

  const float* xb = x + ((size_t)(b * N + src) * Cin) * T + tcl;
  const float* wb = WeT + ((size_t)e * Cout + o) * Cin;

  v8f c = V8F_ZERO;
#pragma unroll 4
  for (int k0 = 0; k0 < Cin; k0 += 4) {
    int k = k0 + (kh << 1);
    v2f a = *(const v2f*)(wb + k);  // A[m][k..k+1], 8B aligned
    v2f bv;
    bv.x = msk * xb[(size_t)k * T];
    bv.y = msk * xb[(size_t)(k + 1) * T];
    c = wmma_k4(a, bv, c);
  }

  float* ab =
      agg + ((size_t)(b * N + dst) * Cout + (ot << 4)) * T + (tt << 4) + m;
  if (t0 < T) {
    for (int r = 0; r < 8; ++r) {
      int mm = r + (kh << 3);
      atomicAdd(&ab[(size_t)mm * T], c[r]);
    }
  }
}

// finalize: h = elu(agg / max(cnt,1) + x @ Wr + bias), in-place on agg
__global__ void finalize_kernel(float* __restrict__ aggh,       // [B,N,Cout,T]
                                const float* __restrict__ cnt,  // [N]
                                const float* __restrict__ hin,  // [B,N,Cin,T]
                                const float* __restrict__ Wr,   // [Cin,Cout]
                                const float* __restrict__ bias,
                                long long total, int N, int Cin, int Cout,
                                int T) {
  long long idx = (long long)blockIdx.x * blockDim.x + threadIdx.x;
  if (idx >= total) return;
  int t = (int)(idx % T);
  int o = (int)((idx / T) % Cout);
  long long bn = idx / ((long long)T * Cout);
  int n = (int)(bn % N);
  const float* hrow = hin + (bn * Cin) * T + t;
  float root = 0.f;
  for (int i = 0; i < Cin; ++i) root += hrow[(size_t)i * T] * Wr[i * Cout + o];
  float cc = cnt[n];
  cc = cc < 1.0f ? 1.0f : cc;
  float v = aggh[idx] / cc + root + bias[o];
  aggh[idx] = v > 0.f ? v : expm1f(v);
}

// Pool GEMM: O[b] = P (M x N) @ H[b] (N x F).  F is a multiple of 64, so each
// wave computes FOUR 16x16 output tiles sharing one A tile (4x A reuse).
__global__ void pool_wmma_kernel(const float* __restrict__ P,  // [M,N]
                                 const float* __restrict__ H,  // [B,N,F]
                                 float* __restrict__ O,        // [B,M,F]
                                 int M, int N, int F, int numTiles) {
  int wave = blockIdx.x * (blockDim.x >> 5) + (threadIdx.x >> 5);
  if (wave >= numTiles) return;
  int lane = threadIdx.x & 31;
  int fQuads = F >> 6;  // groups of four 16-col tiles
  int mTiles = M >> 4;
  int fq = wave % fQuads;
  int tmp = wave / fQuads;
  int mt = tmp % mTiles;
  int b = tmp / mTiles;

  int m = lane & 15;
  int kh = lane >> 4;
  const float* prow = P + (size_t)((mt << 4) + m) * N;        // A[m][*]
  const float* hb = H + (size_t)b * N * F + (fq << 6) + m;    // B[*][n] base

  v8f c0 = V8F_ZERO, c1 = V8F_ZERO, c2 = V8F_ZERO, c3 = V8F_ZERO;
#pragma unroll 2
  for (int k0 = 0; k0 < N; k0 += 4) {
    int k = k0 + (kh << 1);
    v2f a = *(const v2f*)(prow + k);  // contiguous pair, 8B aligned
    __builtin_prefetch(prow + k + 64, 0, 0);  // speculative A-row prefetch
    const float* r0 = hb + (size_t)k * F;
    const float* r1 = hb + (size_t)(k + 1) * F;
    v2f b0, b1, b2, b3;
    b0.x = r0[0];  b0.y = r1[0];
    b1.x = r0[16]; b1.y = r1[16];
    b2.x = r0[32]; b2.y = r1[32];
    b3.x = r0[48]; b3.y = r1[48];
    c0 = wmma_k4(a, b0, c0);
    c1 = wmma_k4(a, b1, c1);
    c2 = wmma_k4(a, b2, c2);
    c3 = wmma_k4(a, b3, c3);
  }
  float* ob = O + ((size_t)b * M + (mt << 4)) * F + (fq << 6) + m;
  for (int r = 0; r < 8; ++r) {
    int mm = r + (kh << 3);
    float* row = ob + (size_t)mm * F;
    row[0] = c0[r];
    row[16] = c1[r];
    row[32] = c2[r];
    row[48] = c3[r];
  }
}

// Channel-mix head: Out[s,o,t] = act(sum_k WT[o,k] * Hin[s,k,t] + bias[o])
// act: 1 = ELU, 2 = tanh
__global__ void chanmix_wmma_kernel(const float* __restrict__ hin,  // [S,Cin,T]
                                    const float* __restrict__ WT,   // [Cout,Cin]
                                    const float* __restrict__ bias,
                                    float* __restrict__ out,  // [S,Cout,T]
                                    int Cin, int Cout, int T, int act,
                                    int numTiles) {
  int wave = blockIdx.x * (blockDim.x >> 5) + (threadIdx.x >> 5);
  if (wave >= numTiles) return;
  int lane = threadIdx.x & 31;
  int tTiles = (T + 15) >> 4;
  int oTiles = Cout >> 4;
  int tt = wave % tTiles;
  int tmp = wave / tTiles;
  int ot = tmp % oTiles;
  int s = tmp / oTiles;

  int m = lane & 15;
  int kh = lane >> 4;
  int o = (ot << 4) + m;
  int t0 = (tt << 4) + m;
  float msk = t0 < T ? 1.0f : 0.0f;
  int tcl = t0 < T ? t0 : T - 1;

  const float* wb = WT + (size_t)o * Cin;
  const float* hb = hin + ((size_t)s * Cin) * T + tcl;

  v8f c = V8F_ZERO;
#pragma unroll 4
  for (int k0 = 0; k0 < Cin; k0 += 4) {
    int k = k0 + (kh << 1);
    v2f a = *(const v2f*)(wb + k);
    v2f bv;
    bv.x = msk * hb[(size_t)k * T];
    bv.y = msk * hb[(size_t)(k + 1) * T];
    c = wmma_k4(a, bv, c);
  }

  float* ob = out + ((size_t)s * Cout + (ot << 4)) * T + (tt << 4) + m;
  if (t0 < T) {
    for (int r = 0; r < 8; ++r) {
      int mm = r + (kh << 3);
      float v = c[r] + bias[(ot << 4) + mm];
      if (act == 1)
        v = v > 0.f ? v : expm1f(v);
      else if (act == 2)
        v = tanhf(v);
      ob[(size_t)mm * T] = v;
    }
  }
}

// ---------------------------------------------------------------- launcher

static inline unsigned nblk(long long n, int bs) {
  return (unsigned)((n + bs - 1) / bs);
}

extern "C" void kernel_launch(void* const* d_in, const int* in_sizes, int n_in,
                              void* d_out, int out_size, void* d_ws,
                              size_t ws_size, hipStream_t stream) {
  (void)in_sizes; (void)n_in; (void)out_size; (void)ws_size;

  const float* x   = (const float*)d_in[0];
  const int*   ei0 = (const int*)d_in[1];
  const float* ea0 = (const float*)d_in[2];
  const int*   ei1 = (const int*)d_in[3];
  const float* ea1 = (const float*)d_in[4];
  const int*   ei2 = (const int*)d_in[5];
  const float* ea2 = (const float*)d_in[6];
  const float* A1  = (const float*)d_in[7];
  const float* Wk1 = (const float*)d_in[8];
  const float* Wr1 = (const float*)d_in[9];
  const float* b1  = (const float*)d_in[10];
  const float* A2  = (const float*)d_in[11];
  const float* Wk2 = (const float*)d_in[12];
  const float* Wr2 = (const float*)d_in[13];
  const float* b2  = (const float*)d_in[14];
  const float* A3  = (const float*)d_in[15];
  const float* Wk3 = (const float*)d_in[16];
  const float* Wr3 = (const float*)d_in[17];
  const float* b3  = (const float*)d_in[18];
  const float* P01 = (const float*)d_in[19];
  const float* P12 = (const float*)d_in[20];
  const float* P23 = (const float*)d_in[21];
  const float* f1w = (const float*)d_in[22];
  const float* f1b = (const float*)d_in[23];
  const float* f2w = (const float*)d_in[24];
  const float* f2b = (const float*)d_in[25];

  constexpr int B = 2, T = 200;
  constexpr int N0 = 2048, N1 = 512, N2 = 128, N3 = 64;
  constexpr int E0 = 16384, E1 = 8192, E2 = 2048;
  constexpr int C0 = 1, C1 = 16, C2 = 32, C3 = 64, C4 = 128, LD = 64;
  constexpr int tT = (T + 15) / 16;  // 13

  // ---- workspace layout (floats) -----------------------------------------
  float* ws = (float*)d_ws;
  size_t off = 0;
  float* We1 = ws + off;   off += (size_t)E0 * C0 * C1;
  float* We2 = ws + off;   off += (size_t)E1 * C1 * C2;
  float* We3 = ws + off;   off += (size_t)E2 * C2 * C3;
  float* f1wT = ws + off;  off += (size_t)C3 * C4;
  float* f2wT = ws + off;  off += (size_t)C4 * LD;
  float* cnt1 = ws + off;  off += N0;
  float* cnt2 = ws + off;  off += N1;
  float* cnt3 = ws + off;  off += N2;
  float* buf0 = ws + off;  off += (size_t)B * N0 * C1 * T;  // agg/h, head tmp
  float* buf1 = ws + off;  off += (size_t)B * N1 * C1 * T;  // pooled h

  // ===================== level 0 -> 1 (Cin=1, Cout=16, N0, E0) ============
  {
    long long na = (long long)B * N0 * C1 * T;
    fill_zero_kernel<<<nblk(na, 256), 256, 0, stream>>>(buf0, na);
    fill_zero_kernel<<<nblk(N0, 256), 256, 0, stream>>>(cnt1, N0);
    edgew_kernel<<<nblk((long long)E0 * C0 * C1, 256), 256, 0, stream>>>(
        ea0, A1, Wk1, We1, E0, C0, C1);
    count_kernel<<<nblk(E0, 256), 256, 0, stream>>>(ei0, cnt1, E0);
    long long nm = (long long)B * E0 * T;
    msg1_kernel<<<nblk(nm, 256), 256, 0, stream>>>(x, We1, ei0, buf0, N0, E0,
                                                   C1, T, nm);
    finalize_kernel<<<nblk(na, 256), 256, 0, stream>>>(buf0, cnt1, x, Wr1, b1,
                                                       na, N0, C0, C1, T);
    int F = C1 * T;  // 3200
    int tiles = B * (N1 / 16) * (F / 64);
    pool_wmma_kernel<<<nblk(tiles, 8), 256, 0, stream>>>(P01, buf0, buf1, N1,
                                                         N0, F, tiles);
  }

  // ===================== level 1 -> 2 (Cin=16, Cout=32, N1, E1) ===========
  {
    long long na = (long long)B * N1 * C2 * T;
    fill_zero_kernel<<<nblk(na, 256), 256, 0, stream>>>(buf0, na);
    fill_zero_kernel<<<nblk(N1, 256), 256, 0, stream>>>(cnt2, N1);
    edgew_kernel<<<nblk((long long)E1 * C1 * C2, 256), 256, 0, stream>>>(
        ea1, A2, Wk2, We2, E1, C1, C2);
    count_kernel<<<nblk(E1, 256), 256, 0, stream>>>(ei1, cnt2, E1);
    int tiles = E1 * B * (C2 / 16) * tT;
    msg_wmma_kernel<<<nblk(tiles, 8), 256, 0, stream>>>(
        buf1, We2, ei1, buf0, N1, E1, C1, C2, B, T, tiles);
    finalize_kernel<<<nblk(na, 256), 256, 0, stream>>>(buf0, cnt2, buf1, Wr2,
                                                       b2, na, N1, C1, C2, T);
    int F = C2 * T;  // 6400
    int ptiles = B * (N2 / 16) * (F / 64);
    pool_wmma_kernel<<<nblk(ptiles, 8), 256, 0, stream>>>(P12, buf0, buf1, N2,
                                                          N1, F, ptiles);
  }

  // ===================== level 2 -> 3 (Cin=32, Cout=64, N2, E2) ===========
  {
    long long na = (long long)B * N2 * C3 * T;
    fill_zero_kernel<<<nblk(na, 256), 256, 0, stream>>>(buf0, na);
    fill_zero_kernel<<<nblk(N2, 256), 256, 0, stream>>>(cnt3, N2);
    edgew_kernel<<<nblk((long long)E2 * C2 * C3, 256), 256, 0, stream>>>(
        ea2, A3, Wk3, We3, E2, C2, C3);
    count_kernel<<<nblk(E2, 256), 256, 0, stream>>>(ei2, cnt3, E2);
    int tiles = E2 * B * (C3 / 16) * tT;
    msg_wmma_kernel<<<nblk(tiles, 8), 256, 0, stream>>>(
        buf1, We3, ei2, buf0, N2, E2, C2, C3, B, T, tiles);
    finalize_kernel<<<nblk(na, 256), 256, 0, stream>>>(buf0, cnt3, buf1, Wr3,
                                                       b3, na, N2, C2, C3, T);
    int F = C3 * T;  // 12800
    int ptiles = B * (N3 / 16) * (F / 64);
    pool_wmma_kernel<<<nblk(ptiles, 8), 256, 0, stream>>>(P23, buf0, buf1, N3,
                                                          N2, F, ptiles);
  }

  // ===================== 1x1 conv head =====================================
  {
    transpose_w_kernel<<<nblk(C3 * C4, 256), 256, 0, stream>>>(f1w, f1wT, C3,
                                                               C4);
    transpose_w_kernel<<<nblk(C4 * LD, 256), 256, 0, stream>>>(f2w, f2wT, C4,
                                                               LD);
    int S = B * N3;  // 128 node/batch slices
    int tiles1 = S * (C4 / 16) * tT;
    chanmix_wmma_kernel<<<nblk(tiles1, 8), 256, 0, stream>>>(
        buf1, f1wT, f1b, buf0, C3, C4, T, /*act=*/1, tiles1);
    int tiles2 = S * (LD / 16) * tT;
    chanmix_wmma_kernel<<<nblk(tiles2, 8), 256, 0, stream>>>(
        buf0, f2wT, f2b, (float*)d_out, C4, LD, T, /*act=*/2, tiles2);
  }
}